// natm_independent_47193100648796
// MI455X (gfx1250) — compile-verified
//
#include <hip/hip_runtime.h>
#include <hip/hip_bf16.h>

// Problem constants (from the reference)
#define B_  4096
#define T_  16
#define F_  48
#define U_  64
#define N_  768        // T*F
#define O_  1

typedef float v2f __attribute__((ext_vector_type(2)));
typedef float v8f __attribute__((ext_vector_type(8)));

// ---------------------------------------------------------------------------
// Kernel 1: weight[b, tf] = sum_u relu(x[b,tf]*w1[tf,u] + b1[tf,u]) * w2[tf,u] + b2[tf]
//
// Block = 256 threads (8 wave32). Lane -> tf within a 32-wide tf tile (coalesced
// x loads / weight stores). Each thread processes 4 batch rows per iteration to
// amortize the LDS parameter reads (12 VALU fma-class ops per 2 DS loads).
// LDS layout [u][32 tf]: float2 (w1,b1) -> banks (2*tfl, 2*tfl+1): conflict-free.
// ---------------------------------------------------------------------------
#define TF_TILE    32
#define BLOCK1     256
#define BPT        4                         // batch rows per thread per iter
#define ROWS_ITER  ((BLOCK1 / TF_TILE) * BPT)   // 32 rows per iteration
#define BATCH_SPAN 256                       // rows handled per block

__global__ __launch_bounds__(BLOCK1)
void natm_feature_kernel(const float* __restrict__ x,
                         const float* __restrict__ w1,
                         const float* __restrict__ b1,
                         const float* __restrict__ w2,
                         const float* __restrict__ b2,
                         float* __restrict__ weight)
{
    __shared__ float2 s_wb[U_ * TF_TILE];   // (w1, b1) pairs, [u][tf] order
    __shared__ float  s_w2[U_ * TF_TILE];   // w2,             [u][tf] order

    const int tf0 = blockIdx.x * TF_TILE;
    const int b0  = blockIdx.y * BATCH_SPAN;
    const int tid = threadIdx.x;

    // Stage parameters for this tf tile into LDS (transposed to [u][tf]).
    for (int e = tid; e < U_ * TF_TILE; e += BLOCK1) {
        const int u   = e >> 5;       // / TF_TILE
        const int tfl = e & 31;       // % TF_TILE
        const int g   = (tf0 + tfl) * U_ + u;
        s_wb[e] = make_float2(w1[g], b1[g]);
        s_w2[e] = w2[g];
    }
    __syncthreads();

    const int tfl  = tid & 31;        // lane -> tf
    const int brow = tid >> 5;        // wave id -> batch sub-row 0..7
    const float b2v = b2[tf0 + tfl];
    const int tf = tf0 + tfl;

    for (int it = 0; it < BATCH_SPAN / ROWS_ITER; ++it) {
        const int bb = b0 + it * ROWS_ITER + brow * BPT;

        float xv[BPT], acc[BPT];
        #pragma unroll
        for (int j = 0; j < BPT; ++j) {
            xv[j]  = x[(size_t)(bb + j) * N_ + tf];   // (x-MEAN)/STD is identity
            acc[j] = 0.0f;
        }

        #pragma unroll 8
        for (int u = 0; u < U_; ++u) {
            const float2 wb = s_wb[u * TF_TILE + tfl];
            const float  wv = s_w2[u * TF_TILE + tfl];
            #pragma unroll
            for (int j = 0; j < BPT; ++j) {
                float h = fmaf(xv[j], wb.x, wb.y);
                h = fmaxf(h, 0.0f);
                acc[j] = fmaf(h, wv, acc[j]);
            }
        }

        #pragma unroll
        for (int j = 0; j < BPT; ++j)
            weight[(size_t)(bb + j) * N_ + tf] = acc[j] + b2v;
    }
}

// ---------------------------------------------------------------------------
// Kernel 2: out[b] = weight[b,:] @ w_out + b_out  via V_WMMA_F32_16X16X4_F32.
//
// One wave per 16-row batch tile; 192 K=4 chunks accumulated into the f32
// accumulator. B-matrix columns are all w_out (broadcast), so every column of
// D holds the same 16 results; lanes with N==0 (lanes 0 and 16) write them out.
// A f32 16x4 layout (ISA 7.12.2): v0 = K=0 (lanes 0-15) / K=2 (lanes 16-31),
// v1 = K=1 / K=3  -> each lane loads one contiguous float2 from its row.
// The same K-slot convention is used to fill B with w_out values.
// ---------------------------------------------------------------------------
#define BLOCK2 256   // 8 waves -> 8 batch tiles per block

__global__ __launch_bounds__(BLOCK2)
void natm_out_kernel(const float* __restrict__ weight,
                     const float* __restrict__ w_out,
                     const float* __restrict__ b_out,
                     float* __restrict__ out)
{
    __shared__ float s_wout[N_];
    for (int e = threadIdx.x; e < N_; e += BLOCK2) s_wout[e] = w_out[e];
    __syncthreads();

    const int wave = threadIdx.x >> 5;
    const int lane = threadIdx.x & 31;
    const int tile = blockIdx.x * (BLOCK2 / 32) + wave;   // 0..255
    const int row  = tile * 16 + (lane & 15);             // M = lane % 16
    const int koff = (lane < 16) ? 0 : 2;                 // K slots per lane half

    const float* __restrict__ arow = weight + (size_t)row * N_;

    v8f c = {0.f, 0.f, 0.f, 0.f, 0.f, 0.f, 0.f, 0.f};

    for (int ch = 0; ch < N_ / 4; ++ch) {
        const int kc = ch * 4 + koff;
        const v2f a = *(const v2f*)(arow + kc);       // 8B-aligned (kc even)
        const v2f b = *(const v2f*)(s_wout + kc);     // broadcast across N
        c = __builtin_amdgcn_wmma_f32_16x16x4_f32(
                /*neg_a=*/false, a, /*neg_b=*/false, b,
                /*c_mod=*/(short)0, c, /*reuse_a=*/false, /*reuse_b=*/false);
    }

    const float bo = b_out[0];
    if ((lane & 15) == 0) {                 // N == 0 columns: lanes 0 and 16
        const int mbase = (lane < 16) ? 0 : 8;
        const int r0 = tile * 16;
        #pragma unroll
        for (int j = 0; j < 8; ++j)
            out[r0 + mbase + j] = c[j] + bo;
    }
}

// ---------------------------------------------------------------------------
// Launcher. d_out = [ out (4096 f32) | weight (4096*768 f32) ] per the
// reference's (out, weight[:, :, None]) tuple, flattened in return order.
// ---------------------------------------------------------------------------
extern "C" void kernel_launch(void* const* d_in, const int* in_sizes, int n_in,
                              void* d_out, int out_size, void* d_ws, size_t ws_size,
                              hipStream_t stream) {
    const float* x     = (const float*)d_in[0];
    const float* w1    = (const float*)d_in[1];
    const float* b1    = (const float*)d_in[2];
    const float* w2    = (const float*)d_in[3];
    const float* b2    = (const float*)d_in[4];
    const float* w_out = (const float*)d_in[5];
    const float* b_out = (const float*)d_in[6];

    float* out    = (float*)d_out;        // [B, O] = 4096 floats
    float* weight = out + B_ * O_;        // [B, N] = 4096*768 floats

    dim3 g1(N_ / TF_TILE, B_ / BATCH_SPAN);   // 24 x 16 = 384 blocks
    natm_feature_kernel<<<g1, BLOCK1, 0, stream>>>(x, w1, b1, w2, b2, weight);

    dim3 g2((B_ / 16) / (BLOCK2 / 32));       // 256 tiles / 8 per block = 32 blocks
    natm_out_kernel<<<g2, BLOCK2, 0, stream>>>(weight, w_out, b_out, out);
}